// Loss_35974646071982
// MI455X (gfx1250) — compile-verified
//
#include <hip/hip_runtime.h>
#include <hip/hip_bf16.h>
#include <math.h>

typedef __attribute__((ext_vector_type(2))) float v2f;
typedef __attribute__((ext_vector_type(8))) float v8f;

#define BB 2
#define TT 5
#define HH 64
#define WW 2048
#define NP 4096                    // N == M == 4096 sampled points
#define BT (BB*TT)                 // 10 (b,t) pairs
#define RV_COUNT (BB*TT*HH*WW)     // 1,310,720 range-image elements

// ---------------------------------------------------------------------------
// Workspace layout (floats): [0..9] sum(dist1) per bt, [10..19] sum(dist2),
// [20] rv L1 sum, [21] mask BCE sum.
// ---------------------------------------------------------------------------

__global__ void init_acc_kernel(float* acc) {
    int i = threadIdx.x;
    if (i < 22) acc[i] = 0.0f;
}

// ---------------------------------------------------------------------------
// Chamfer one-direction kernel: for each query row n, min over all m of
//   d2(n,m) = |p_n|^2 + |q_m|^2 - 2 p_n . q_m
// One V_WMMA_F32_16X16X4_F32 produces a complete 16x16 tile of d2:
//   A[n][*] = (px, py, pz, 1)                    (loop-invariant, 2 VGPRs)
//   B[*][m] = (-2qx, -2qy, -2qz, |q|^2)          (one ds_load_b128 per tile;
//                                                 pre-scaled at LDS-fill time)
//   C[n][m] = |p|^2 per row                      (loop-invariant, built once
//                                                 with 8 lane shuffles)
// Block = 256 threads (8 waves); each wave owns 16 n-rows; the whole 4096-pt
// database side lives in LDS as float4 (64 KB; WGP has 320 KB).
// ---------------------------------------------------------------------------
__global__ __launch_bounds__(256)
void chamfer_min_kernel(const float* __restrict__ Pn,   // query points  [BT,NP,3]
                        const float* __restrict__ Qm,   // database pts  [BT,NP,3]
                        float* __restrict__ acc)        // per-bt sum of row mins
{
    extern __shared__ float4 q4[];                      // [NP] = 64 KB

    const int bt  = blockIdx.y;
    const int tid = threadIdx.x;
    const float* Q = Qm + (size_t)bt * NP * 3;

    // Cooperative LDS fill: pre-scaled coordinates + squared norm, packed.
    for (int i = tid; i < NP; i += 256) {
        float qx = Q[i*3 + 0], qy = Q[i*3 + 1], qz = Q[i*3 + 2];
        q4[i] = make_float4(-2.0f*qx, -2.0f*qy, -2.0f*qz, qx*qx + qy*qy + qz*qz);
    }
    __syncthreads();

    const int wave = tid >> 5;
    const int lane = tid & 31;
    const int half = lane >> 4;    // 0: lanes 0-15 carry K=0,1 ; 1: lanes 16-31 carry K=2,3
    const int l16  = lane & 15;

    // This wave's 16 query rows; lane-half selects which K-pair it carries.
    const int n = blockIdx.x * 128 + wave * 16 + l16;
    const float* P = Pn + ((size_t)bt * NP + n) * 3;
    const float px = P[0], py = P[1], pz = P[2];
    const float pn = px*px + py*py + pz*pz;

    v2f a;                         // A-matrix 16x4 f32: row n = (x, y, z, 1)
    a.x = half ? pz   : px;        // v0: K=0 (x) | K=2 (z)
    a.y = half ? 1.0f : py;        // v1: K=1 (y) | K=3 (1)

    // Loop-invariant C: C[row][col] = |p_row|^2. VGPR r, lane l holds row
    // (r + 8*(l>=16)); row j's |p|^2 lives in lanes j and j+16, so shuffle
    // from lane (r + 8*half).
    v8f con;
    #pragma unroll
    for (int r = 0; r < 8; ++r) con[r] = __shfl(pn, r + 8*half, 32);

    v8f minv;
    #pragma unroll
    for (int r = 0; r < 8; ++r) minv[r] = 3.4e38f;

    for (int mb = 0; mb < NP; mb += 16) {
        // One b128 LDS read per lane: this lane's column (both halves read the
        // same 16 points -> LDS broadcast; 16-B stride is bank-conflict-free).
        const float4 q = q4[mb + l16];

        v2f b;                     // B-matrix 4x16 f32: col m = (-2q, |q|^2)
        b.x = half ? q.y : q.x;    // v0: K=0 row | K=1 row
        b.y = half ? q.w : q.z;    // v1: K=2 row | K=3 row

        // D = A x B + C  ==> full 16x16 tile of squared distances.
        v8f d = __builtin_amdgcn_wmma_f32_16x16x4_f32(
                    false, a, false, b, (short)0, con, false, false);

        #pragma unroll
        for (int r = 0; r < 8; ++r) minv[r] = fminf(minv[r], d[r]);
    }

    // Min across the 16 columns (lanes of each half). D layout: VGPR r, lane l
    // holds row (r + 8*(l>=16)), col l&15.
    #pragma unroll
    for (int off = 1; off < 16; off <<= 1) {
        #pragma unroll
        for (int r = 0; r < 8; ++r)
            minv[r] = fminf(minv[r], __shfl_xor(minv[r], off, 32));
    }
    // Sum this wave's 16 per-row minima (rows 0-7 in low half, 8-15 in high half).
    float s = 0.0f;
    #pragma unroll
    for (int r = 0; r < 8; ++r) s += minv[r];
    s += __shfl_xor(s, 16, 32);
    if (lane == 0) atomicAdd(&acc[bt], s);
}

// ---------------------------------------------------------------------------
// Fused L1-range + BCE-with-logits reduction (bandwidth-bound, ~15.7 MB).
// ---------------------------------------------------------------------------
__global__ __launch_bounds__(256)
void rv_mask_kernel(const float* __restrict__ rv, const float* __restrict__ ml,
                    const float* __restrict__ tg, float* __restrict__ acc)
{
    __shared__ float s_rv[8], s_mk[8];
    float srv = 0.0f, smk = 0.0f;
    for (int i = blockIdx.x * blockDim.x + threadIdx.x; i < RV_COUNT;
         i += gridDim.x * blockDim.x) {
        float t = tg[i];
        float o = (t == -1.0f) ? -1.0f : rv[i];
        srv += fabsf(o - t);
        float x  = ml[i];
        float tm = (t > 0.0f) ? 1.0f : 0.0f;
        smk += fmaxf(x, 0.0f) - x * tm + log1pf(expf(-fabsf(x)));
    }
    #pragma unroll
    for (int off = 16; off >= 1; off >>= 1) {
        srv += __shfl_xor(srv, off, 32);
        smk += __shfl_xor(smk, off, 32);
    }
    const int lane = threadIdx.x & 31, wave = threadIdx.x >> 5;
    if (lane == 0) { s_rv[wave] = srv; s_mk[wave] = smk; }
    __syncthreads();
    if (threadIdx.x == 0) {
        float a = 0.0f, b = 0.0f;
        #pragma unroll
        for (int w = 0; w < 8; ++w) { a += s_rv[w]; b += s_mk[w]; }
        atomicAdd(&acc[20], a);
        atomicAdd(&acc[21], b);
    }
}

// ---------------------------------------------------------------------------
// Finalize: combine sums -> (total, cd_tensor[T,B], loss_cd, loss_rv, loss_mask)
// ---------------------------------------------------------------------------
__global__ void finalize_kernel(const float* __restrict__ acc, float* __restrict__ out)
{
    if (blockIdx.x == 0 && threadIdx.x == 0) {
        const float inv_rv = 1.0f / (float)RV_COUNT;
        const float lrv = acc[20] * inv_rv;
        const float lmk = acc[21] * inv_rv;
        float cdbt[BT];
        float lcd = 0.0f;
        for (int b = 0; b < BB; ++b)
            for (int t = 0; t < TT; ++t) {
                const int i = b * TT + t;
                const float cd = (acc[i] + acc[10 + i]) * (1.0f / (float)NP);
                cdbt[i] = cd;
                lcd += cd;
            }
        lcd *= (1.0f / (float)BT);
        out[0] = lcd + lrv + lmk;                 // total (W_CD=W_RV=W_MASK=1)
        for (int t = 0; t < TT; ++t)              // cd_tensor is [T,B] (transposed)
            for (int b = 0; b < BB; ++b)
                out[1 + t * BB + b] = cdbt[b * TT + t];
        out[11] = lcd;
        out[12] = lrv;
        out[13] = lmk;
    }
}

extern "C" void kernel_launch(void* const* d_in, const int* in_sizes, int n_in,
                              void* d_out, int out_size, void* d_ws, size_t ws_size,
                              hipStream_t stream) {
    const float* output_rv       = (const float*)d_in[0];
    const float* mask_logits     = (const float*)d_in[1];
    const float* output_points   = (const float*)d_in[2];
    const float* target_points   = (const float*)d_in[3];
    const float* target_range    = (const float*)d_in[4];
    float* out = (float*)d_out;
    float* acc = (float*)d_ws;

    init_acc_kernel<<<1, 32, 0, stream>>>(acc);

    rv_mask_kernel<<<1024, 256, 0, stream>>>(output_rv, mask_logits, target_range, acc);

    const size_t lds_bytes = NP * sizeof(float4);      // 64 KB (WGP has 320 KB)
    dim3 grid(NP / 128, BT);                           // 32 row-tiles x 10 (b,t)
    // dist1: for each output point, nearest target point
    chamfer_min_kernel<<<grid, 256, lds_bytes, stream>>>(output_points, target_points, acc + 0);
    // dist2: for each target point, nearest output point
    chamfer_min_kernel<<<grid, 256, lds_bytes, stream>>>(target_points, output_points, acc + 10);

    finalize_kernel<<<1, 32, 0, stream>>>(acc, out);
}